// TransformerBlock_20177756356855
// MI455X (gfx1250) — compile-verified
//
#include <hip/hip_runtime.h>
#include <hip/hip_bf16.h>

// ---------------------------------------------------------------------------
// Restormer-style transformer block on [8,192,64,64] fp32 for gfx1250.
// All GEMM-shaped work (1x1 convs, q.k^T, attn.v) runs on V_WMMA_F32_16X16X4_F32
// (full fp32 precision; the block is HBM-bandwidth-bound, so fp32 WMMA costs
// nothing vs f16 while avoiding accuracy risk).
// Hot loops are guard-free: K-tails handled by zero-padding in workspace,
// M-tails by clamped (always in-bounds) loads + guarded stores only.
// ---------------------------------------------------------------------------

typedef float v2f __attribute__((ext_vector_type(2)));
typedef float v8f __attribute__((ext_vector_type(8)));

#define BB    8
#define CDIM  192
#define HW    4096
#define NPIX  (BB * HW)
#define HEADS 4
#define CH    48          // channels per head
#define QKVC  576
#define HID   510
#define HIDP  512         // padded hidden for guard-free K loop
#define FFNC  1020

__device__ __forceinline__ v8f wmma4(v2f a, v2f b, v8f c) {
  // D = A(16x4 f32) * B(4x16 f32) + C(16x16 f32)
  return __builtin_amdgcn_wmma_f32_16x16x4_f32(false, a, false, b, (short)0, c,
                                               false, false);
}

// ---------------- LayerNorm over channel axis (per pixel) -------------------
__global__ __launch_bounds__(256)
void ln_c_kernel(const float* __restrict__ X, const float* __restrict__ w,
                 const float* __restrict__ bias, float* __restrict__ Y) {
  int pix = blockIdx.x * 256 + threadIdx.x;          // 0 .. NPIX-1
  int b = pix / HW, n = pix % HW;
  const float* xp = X + (size_t)b * CDIM * HW + n;
  float s = 0.f, sq = 0.f;
  for (int c = 0; c < CDIM; ++c) {
    float v = xp[(size_t)c * HW];
    s += v; sq += v * v;
  }
  float mu = s * (1.f / CDIM);
  float var = sq * (1.f / CDIM) - mu * mu;
  float inv = rsqrtf(var + 1e-5f);
  float* yp = Y + (size_t)b * CDIM * HW + n;
  for (int c = 0; c < CDIM; ++c) {
    float v = xp[(size_t)c * HW];
    yp[(size_t)c * HW] = (v - mu) * inv * w[c] + bias[c];
  }
}

// ---------------- 1x1 conv as WMMA GEMM:  Y[b,o,n] = sum_c W[o,c] X[b,c,n] --
// block = 128 threads (4 waves), block tile M=64 (16 per wave) x N=64.
// K must be a multiple of 4 (pad in workspace if needed). No guards in K loop.
__global__ __launch_bounds__(128)
void gemm_wmma(const float* __restrict__ Wt,   // [O, K]
               const float* __restrict__ X,    // [B, K, HW]
               const float* __restrict__ Res,  // [B, O, HW] or nullptr
               float* __restrict__ Y,          // [B, O, HW]
               int O, int K) {
  const int lane = threadIdx.x & 31;
  const int wave = threadIdx.x >> 5;
  const int lane16 = lane & 15;
  const int half = lane >> 4;
  const int b = blockIdx.z;
  const int m0 = blockIdx.y * 64 + wave * 16;
  const int n0 = blockIdx.x * 64;
  const float* Xb = X + (size_t)b * K * HW + n0 + lane16;
  const int row = min(m0 + lane16, O - 1);           // clamp: always in-bounds
  const float* Wp = Wt + (size_t)row * K;
  v8f acc0 = {}, acc1 = {}, acc2 = {}, acc3 = {};

  for (int k0 = 0; k0 < K; k0 += 4) {
    const int ka = k0 + 2 * half;
    v2f a = *(const v2f*)(Wp + ka);                  // 8B aligned (ka even)
    const float* xp = Xb + (size_t)ka * HW;
    v2f b0, b1, b2, b3;
    b0.x = xp[0];  b0.y = xp[HW];
    b1.x = xp[16]; b1.y = xp[HW + 16];
    b2.x = xp[32]; b2.y = xp[HW + 32];
    b3.x = xp[48]; b3.y = xp[HW + 48];
    acc0 = wmma4(a, b0, acc0);
    acc1 = wmma4(a, b1, acc1);
    acc2 = wmma4(a, b2, acc2);
    acc3 = wmma4(a, b3, acc3);
  }

  if (Res) {
#pragma unroll
    for (int r = 0; r < 8; ++r) {
      int o = m0 + r + 8 * half;
      if (o < O) {
        size_t base = ((size_t)b * O + o) * HW + n0 + lane16;
        Y[base]      = acc0[r] + Res[base];
        Y[base + 16] = acc1[r] + Res[base + 16];
        Y[base + 32] = acc2[r] + Res[base + 32];
        Y[base + 48] = acc3[r] + Res[base + 48];
      }
    }
  } else {
#pragma unroll
    for (int r = 0; r < 8; ++r) {
      int o = m0 + r + 8 * half;
      if (o < O) {
        size_t base = ((size_t)b * O + o) * HW + n0 + lane16;
        Y[base]      = acc0[r];
        Y[base + 16] = acc1[r];
        Y[base + 32] = acc2[r];
        Y[base + 48] = acc3[r];
      }
    }
  }
}

// ---------------- depthwise 3x3, pad 1 --------------------------------------
__global__ __launch_bounds__(256)
void dwconv3_kernel(const float* __restrict__ X, const float* __restrict__ Wd,
                    float* __restrict__ Y, int NCH) {
  size_t idx = (size_t)blockIdx.x * 256 + threadIdx.x;
  int n = (int)(idx % HW);
  int c = (int)((idx / HW) % NCH);
  int y = n >> 6, x = n & 63;
  const float* xp = X + (idx - n);      // start of this [64x64] plane
  const float* wp = Wd + (size_t)c * 9;
  float acc = 0.f;
#pragma unroll
  for (int dy = 0; dy < 3; ++dy) {
#pragma unroll
    for (int dx = 0; dx < 3; ++dx) {
      int yy = y + dy - 1, xx = x + dx - 1;
      if (yy >= 0 && yy < 64 && xx >= 0 && xx < 64)
        acc += xp[yy * 64 + xx] * wp[dy * 3 + dx];
    }
  }
  Y[idx] = acc;
}

// ---------------- L2 row norms for q and k (per head-row over HW) -----------
__global__ __launch_bounds__(256)
void rownorm_kernel(const float* __restrict__ qkv, float* __restrict__ invq,
                    float* __restrict__ invk) {
  int wave = threadIdx.x >> 5, lane = threadIdx.x & 31;
  int row = blockIdx.x * 8 + wave;           // 0 .. B*HEADS*96-1
  int bh = row / 96, within = row % 96;
  int b = bh >> 2, h = bh & 3;
  int isk = within >= 48 ? 1 : 0;
  int c = within - isk * 48;
  const float* P = qkv + ((size_t)b * QKVC + isk * CDIM + h * CH + c) * HW;
  float s = 0.f;
  for (int i = lane; i < HW; i += 32) { float v = P[i]; s += v * v; }
#pragma unroll
  for (int m = 16; m; m >>= 1) s += __shfl_xor(s, m, 32);
  if (lane == 0) {
    float inv = 1.f / fmaxf(sqrtf(s), 1e-12f);
    (isk ? invk : invq)[bh * 48 + c] = inv;
  }
}

// ---------------- attn logits: (48xHW).(HWx48) via WMMA, one wave per (b,h) -
__global__ __launch_bounds__(32)
void attn_qk_kernel(const float* __restrict__ qkv, const float* __restrict__ invq,
                    const float* __restrict__ invk, const float* __restrict__ temp,
                    float* __restrict__ logits) {
  const int b = blockIdx.x >> 2, h = blockIdx.x & 3;
  const int lane = threadIdx.x;
  const int lane16 = lane & 15, half = lane >> 4;
  const float* Q  = qkv + ((size_t)b * QKVC + h * CH) * HW;
  const float* Kp = qkv + ((size_t)b * QKVC + CDIM + h * CH) * HW;
  v8f acc[3][3] = {};

  for (int k0 = 0; k0 < HW; k0 += 4) {
    const int kk = k0 + 2 * half;
    v2f a[3], bb[3];
#pragma unroll
    for (int t = 0; t < 3; ++t) {
      size_t r = (size_t)(t * 16 + lane16) * HW + kk;
      a[t]  = *(const v2f*)(Q + r);       // 8B aligned (kk even)
      bb[t] = *(const v2f*)(Kp + r);
    }
#pragma unroll
    for (int mt = 0; mt < 3; ++mt)
#pragma unroll
      for (int nt = 0; nt < 3; ++nt)
        acc[mt][nt] = wmma4(a[mt], bb[nt], acc[mt][nt]);
  }

  const int bh = b * 4 + h;
  const float tv = temp[h];
#pragma unroll
  for (int mt = 0; mt < 3; ++mt)
#pragma unroll
    for (int nt = 0; nt < 3; ++nt)
#pragma unroll
      for (int r = 0; r < 8; ++r) {
        int c = mt * 16 + r + 8 * half;
        int d = nt * 16 + lane16;
        logits[((size_t)bh * 48 + c) * 48 + d] =
            acc[mt][nt][r] * invq[bh * 48 + c] * invk[bh * 48 + d] * tv;
      }
}

// ---------------- softmax over last dim (48), one thread per row ------------
__global__ __launch_bounds__(256)
void softmax48_kernel(float* __restrict__ logits) {
  int row = blockIdx.x * 256 + threadIdx.x;      // 0 .. B*HEADS*48-1
  if (row >= BB * HEADS * 48) return;
  float* p = logits + (size_t)row * 48;
  float m = -3.402823e38f;
  for (int i = 0; i < 48; ++i) m = fmaxf(m, p[i]);
  float s = 0.f;
  for (int i = 0; i < 48; ++i) { float e = __expf(p[i] - m); p[i] = e; s += e; }
  float inv = 1.f / s;
  for (int i = 0; i < 48; ++i) p[i] *= inv;
}

// ---------------- out = attn(48x48, LDS) . v(48xHW) via WMMA ----------------
__global__ __launch_bounds__(256)
void attn_av_kernel(const float* __restrict__ attn, const float* __restrict__ qkv,
                    float* __restrict__ Outp) {
  __shared__ float As[48 * 48];
  const int b = blockIdx.x >> 2, h = blockIdx.x & 3;
  const int tid = threadIdx.x;
  const float* Ab = attn + (size_t)(b * 4 + h) * 48 * 48;
  for (int i = tid; i < 48 * 48; i += 256) As[i] = Ab[i];
  __syncthreads();

  const int lane = tid & 31, wave = tid >> 5;
  const int lane16 = lane & 15, half = lane >> 4;
  const float* V = qkv + ((size_t)b * QKVC + 2 * CDIM + h * CH) * HW;
  float* Op = Outp + ((size_t)b * CDIM + h * CH) * HW;

  for (int nt = wave; nt < HW / 16; nt += 8) {
    const int n = nt * 16 + lane16;
    v8f acc0 = {}, acc1 = {}, acc2 = {};
    for (int d0 = 0; d0 < 48; d0 += 4) {
      const int kk = d0 + 2 * half;
      v2f a0 = *(const v2f*)(&As[(lane16)      * 48 + kk]);   // ds_load_b64
      v2f a1 = *(const v2f*)(&As[(16 + lane16) * 48 + kk]);
      v2f a2 = *(const v2f*)(&As[(32 + lane16) * 48 + kk]);
      v2f bf;
      bf.x = V[(size_t)kk * HW + n];
      bf.y = V[(size_t)(kk + 1) * HW + n];
      acc0 = wmma4(a0, bf, acc0);
      acc1 = wmma4(a1, bf, acc1);
      acc2 = wmma4(a2, bf, acc2);
    }
#pragma unroll
    for (int r = 0; r < 8; ++r) {
      Op[(size_t)(r + 8 * half) * HW + n]      = acc0[r];
      Op[(size_t)(16 + r + 8 * half) * HW + n] = acc1[r];
      Op[(size_t)(32 + r + 8 * half) * HW + n] = acc2[r];
    }
  }
}

// ------- gated = gelu(x1) * x2 into padded [B, 512, HW] (chs 510/511 = 0) ---
__global__ __launch_bounds__(256)
void gate_kernel(const float* __restrict__ Hd, float* __restrict__ G) {
  size_t idx = (size_t)blockIdx.x * 256 + threadIdx.x;   // B*HIDP*HW elems
  int n = (int)(idx % HW);
  int c = (int)((idx / HW) % HIDP);
  int b = (int)(idx / ((size_t)HIDP * HW));
  float r = 0.f;
  if (c < HID) {                                          // uniform per block
    float x1 = Hd[((size_t)b * FFNC + c) * HW + n];
    float x2 = Hd[((size_t)b * FFNC + HID + c) * HW + n];
    float t = 0.7978845608028654f * (x1 + 0.044715f * x1 * x1 * x1);
    r = 0.5f * x1 * (1.f + tanhf(t)) * x2;
  }
  G[idx] = r;
}

// ------- zero-pad ffn_out weight [192,510] -> [192,512] ---------------------
__global__ __launch_bounds__(256)
void padw_kernel(const float* __restrict__ W, float* __restrict__ Wp) {
  int idx = blockIdx.x * 256 + threadIdx.x;              // CDIM*HIDP elems
  int o = idx / HIDP, k = idx % HIDP;
  Wp[idx] = (k < HID) ? W[o * HID + k] : 0.f;
}

// ---------------------------------------------------------------------------
extern "C" void kernel_launch(void* const* d_in, const int* in_sizes, int n_in,
                              void* d_out, int out_size, void* d_ws, size_t ws_size,
                              hipStream_t stream) {
  const float* x       = (const float*)d_in[0];
  const float* ln1w    = (const float*)d_in[1];
  const float* ln1b    = (const float*)d_in[2];
  const float* temp    = (const float*)d_in[3];
  const float* qkvw    = (const float*)d_in[4];
  const float* qkvdw   = (const float*)d_in[5];
  const float* projw   = (const float*)d_in[6];
  const float* ln2w    = (const float*)d_in[7];
  const float* ln2b    = (const float*)d_in[8];
  const float* ffninw  = (const float*)d_in[9];
  const float* ffndw   = (const float*)d_in[10];
  const float* ffnoutw = (const float*)d_in[11];
  float* out = (float*)d_out;

  float* ws = (float*)d_ws;
  float* R0   = ws;                                  // [B,192,HW]  xn
  float* R1   = R0 + (size_t)BB * CDIM * HW;         // [B,1020,HW] pre-dw
  float* R2   = R1 + (size_t)BB * FFNC * HW;         // [B,1020,HW] post-dw
  float* R3   = R2 + (size_t)BB * FFNC * HW;         // [B,512,HW]  attn_out/gated
  float* R4   = R3 + (size_t)BB * HIDP * HW;         // [B,192,HW]  x after attn
  float* INVQ = R4 + (size_t)BB * CDIM * HW;         // [B*H*48]
  float* INVK = INVQ + BB * HEADS * 48;
  float* LOG  = INVK + BB * HEADS * 48;              // [B*H*48*48]
  float* WPAD = LOG + BB * HEADS * 48 * 48;          // [192,512]

  // --- attention branch ---
  ln_c_kernel<<<NPIX / 256, 256, 0, stream>>>(x, ln1w, ln1b, R0);
  gemm_wmma<<<dim3(HW / 64, QKVC / 64, BB), 128, 0, stream>>>(qkvw, R0, nullptr,
                                                              R1, QKVC, CDIM);
  dwconv3_kernel<<<(BB * QKVC * HW) / 256, 256, 0, stream>>>(R1, qkvdw, R2, QKVC);
  rownorm_kernel<<<(BB * HEADS * 96) / 8, 256, 0, stream>>>(R2, INVQ, INVK);
  attn_qk_kernel<<<BB * HEADS, 32, 0, stream>>>(R2, INVQ, INVK, temp, LOG);
  softmax48_kernel<<<(BB * HEADS * 48 + 255) / 256, 256, 0, stream>>>(LOG);
  attn_av_kernel<<<BB * HEADS, 256, 0, stream>>>(LOG, R2, R3);
  gemm_wmma<<<dim3(HW / 64, CDIM / 64, BB), 128, 0, stream>>>(projw, R3, x, R4,
                                                              CDIM, CDIM);
  // --- FFN branch ---
  ln_c_kernel<<<NPIX / 256, 256, 0, stream>>>(R4, ln2w, ln2b, R0);
  gemm_wmma<<<dim3(HW / 64, (FFNC + 63) / 64, BB), 128, 0, stream>>>(
      ffninw, R0, nullptr, R1, FFNC, CDIM);
  dwconv3_kernel<<<(BB * FFNC * HW) / 256, 256, 0, stream>>>(R1, ffndw, R2, FFNC);
  padw_kernel<<<(CDIM * HIDP) / 256, 256, 0, stream>>>(ffnoutw, WPAD);
  gate_kernel<<<(BB * HIDP * HW) / 256, 256, 0, stream>>>(R2, R3);
  gemm_wmma<<<dim3(HW / 64, CDIM / 64, BB), 128, 0, stream>>>(WPAD, R3, R4,
                                                              out, CDIM, HIDP);
}